// F1_66365834657892
// MI455X (gfx1250) — compile-verified
//
#include <hip/hip_runtime.h>

#define N_SAMPLES   (1u << 20)
#define N_CLASSES   128
#define TILE_ROWS   8                         // rows per per-wave tile
#define TILE_ELEMS  (TILE_ROWS * N_CLASSES)   // 1024 floats = 4 KB
#define NUM_TILES   (N_SAMPLES / TILE_ROWS)   // 131072
#define MAIN_BLOCKS 2048
#define WAVES_PER_BLOCK 8
#define F1_EPS 1e-7f

typedef __attribute__((ext_vector_type(4))) unsigned int v4u_t;
typedef __attribute__((ext_vector_type(8))) int          v8i_t;
typedef __attribute__((ext_vector_type(4))) int          v4i_t;

#if __has_include(<hip/amd_detail/amd_gfx1250_TDM.h>)
#define TDM_HAS_6ARG 1
#endif

// Issue a TDM DMA of one contiguous 1024-float tile (global -> LDS).
// D# group0: count=1 | lds_addr | global_addr[56:0] | type=2
// D# group1: data_size=4B, 1-D tile: tensor_dim0=tile_dim0=1024,
//            tensor_dim1=tile_dim1=1, tensor_dim0_stride=1024.
__device__ __forceinline__ void tdm_load_tile_async(const float* gsrc,
                                                    unsigned int lds_byte_off) {
  unsigned long long ga = (unsigned long long)(size_t)gsrc;
  v4u_t g0;
  g0.x = 1u;                                                    // count=1 (valid)
  g0.y = lds_byte_off;                                          // lds_addr (bytes)
  g0.z = (unsigned int)ga;                                      // global_addr[31:0]
  g0.w = ((unsigned int)(ga >> 32) & 0x01FFFFFFu) | (2u << 30); // addr[56:32] | type=2
  v8i_t g1;
  g1.s0 = (int)(2u << 16);                                      // data_size=2 -> 4B; wg_mask=0
  g1.s1 = (int)(((unsigned)TILE_ELEMS & 0xFFFFu) << 16);        // tensor_dim0[15:0]
  g1.s2 = (int)(((unsigned)TILE_ELEMS >> 16) | (1u << 16));     // tensor_dim0 hi | tensor_dim1=1
  g1.s3 = (int)((unsigned)TILE_ELEMS << 16);                    // tile_dim0
  g1.s4 = 1;                                                    // tile_dim1=1, tile_dim2=0
  g1.s5 = (int)TILE_ELEMS;                                      // tensor_dim0_stride
  g1.s6 = 0;
  g1.s7 = 0;
  v4i_t z4 = {0, 0, 0, 0};
#ifdef TDM_HAS_6ARG
  v8i_t z8 = {0, 0, 0, 0, 0, 0, 0, 0};
  __builtin_amdgcn_tensor_load_to_lds(g0, g1, z4, z4, z8, 0);
#else
  __builtin_amdgcn_tensor_load_to_lds(g0, g1, z4, z4, 0);
#endif
}

__global__ void f1_zero_cm(unsigned int* __restrict__ cm) {
  const int i = blockIdx.x * blockDim.x + threadIdx.x;
  if (i < N_CLASSES * N_CLASSES) cm[i] = 0u;
}

// Streaming argmax + confusion-matrix histogram.
// Each wave owns a private double-buffered TDM stream (TENSORcnt is per-wave),
// so the steady-state loop has NO workgroup barriers: issue next tile's DMA,
// s_wait_tensorcnt(1) for the current one, consume 8 rows from LDS.
__global__ __launch_bounds__(256, 2) void f1_confusion_kernel(
    const float* __restrict__ y_pred, const int* __restrict__ y_true,
    unsigned int* __restrict__ cm) {
  __shared__ unsigned int s_hist[N_CLASSES * N_CLASSES];                   // 64 KB
  __shared__ __align__(16) float s_buf[WAVES_PER_BLOCK][2][TILE_ELEMS];    // 64 KB

  const int tid  = (int)threadIdx.x;
  const int lane = tid & 31;
  const int wave = tid >> 5;

  for (int i = tid; i < N_CLASSES * N_CLASSES; i += 256) s_hist[i] = 0u;

  int tile = (int)blockIdx.x * WAVES_PER_BLOCK + wave;   // global wave id
  const int tstride = (int)gridDim.x * WAVES_PER_BLOCK;

  // Prologue: start this wave's first DMA before the histogram barrier so the
  // fill overlaps the LDS zeroing.
  if (tile < NUM_TILES) {
    tdm_load_tile_async(y_pred + (size_t)tile * TILE_ELEMS,
                        (unsigned int)(size_t)&s_buf[wave][0][0]);
  }
  __syncthreads();  // s_hist zeroed (TDM targets s_buf only; no conflict)

  int b = 0;
  for (; tile < NUM_TILES; tile += tstride, b ^= 1) {
    const int nxt = tile + tstride;
    if (nxt < NUM_TILES) {
      // Safe overwrite: buf[b^1]'s previous contents were fully consumed last
      // iteration (its ds_load results were waited on before use), and same-wave
      // tensor ops complete in order.
      tdm_load_tile_async(y_pred + (size_t)nxt * TILE_ELEMS,
                          (unsigned int)(size_t)&s_buf[wave][b ^ 1][0]);
      __builtin_amdgcn_s_wait_tensorcnt(1);  // current tile has landed
    } else {
      __builtin_amdgcn_s_wait_tensorcnt(0);
    }

    const float* buf = &s_buf[wave][b][0];
#pragma unroll
    for (int r = 0; r < TILE_ROWS; ++r) {
      // One wave reads a full 512 B row: lane l covers columns 4l..4l+3.
      const float4 v = *(const float4*)(buf + r * N_CLASSES + lane * 4);
      float mv = v.x;
      int   mi = lane * 4;
      if (v.y > mv) { mv = v.y; mi = lane * 4 + 1; }
      if (v.z > mv) { mv = v.z; mi = lane * 4 + 2; }
      if (v.w > mv) { mv = v.w; mi = lane * 4 + 3; }
      // Wave32 butterfly argmax; ties -> smallest index (jnp.argmax semantics).
#pragma unroll
      for (int off = 16; off >= 1; off >>= 1) {
        const float ov = __shfl_xor(mv, off, 32);
        const int   oi = __shfl_xor(mi, off, 32);
        if (ov > mv || (ov == mv && oi < mi)) { mv = ov; mi = oi; }
      }
      if (lane == 0) {
        const int t = y_true[(size_t)tile * TILE_ROWS + r];
        atomicAdd(&s_hist[t * N_CLASSES + mi], 1u);
      }
    }
  }

  __syncthreads();  // all waves' histogram updates done
  for (int i = tid; i < N_CLASSES * N_CLASSES; i += 256) {
    const unsigned int v = s_hist[i];
    if (v) atomicAdd(&cm[i], v);
  }
}

// Exact reference formulas (incl. its integer fancy-indexing for FP/FN).
__global__ void f1_finalize(const unsigned int* __restrict__ cm,
                            float* __restrict__ out) {
  __shared__ float s_sens[N_CLASSES];
  __shared__ float s_prec[N_CLASSES];
  const int c = (int)threadIdx.x;
  const float TP = (float)cm[c * N_CLASSES + c];
  const float FP = (float)(N_CLASSES - 1) * (float)cm[c * N_CLASSES + 1] +
                   (float)cm[c * N_CLASSES + 0];
  const float FN = (float)(N_CLASSES - 1) * (float)cm[1 * N_CLASSES + c] +
                   (float)cm[0 * N_CLASSES + c];
  s_sens[c] = TP / (TP + FN + F1_EPS);
  s_prec[c] = TP / (TP + FP + F1_EPS);
  __syncthreads();
  for (int off = N_CLASSES / 2; off > 0; off >>= 1) {
    if (c < off) {
      s_sens[c] += s_sens[c + off];
      s_prec[c] += s_prec[c + off];
    }
    __syncthreads();
  }
  if (c == 0) {
    const float sens = s_sens[0] * (1.0f / (float)N_CLASSES);
    const float prec = s_prec[0] * (1.0f / (float)N_CLASSES);
    out[0] = 2.0f * (prec * sens / (prec + sens + F1_EPS));
  }
}

extern "C" void kernel_launch(void* const* d_in, const int* in_sizes, int n_in,
                              void* d_out, int out_size, void* d_ws, size_t ws_size,
                              hipStream_t stream) {
  (void)in_sizes; (void)n_in; (void)out_size; (void)ws_size;
  const float* y_pred = (const float*)d_in[0];
  const int*   y_true = (const int*)d_in[1];
  float*        out   = (float*)d_out;
  unsigned int* cm    = (unsigned int*)d_ws;  // 128*128*4 = 64 KB of workspace

  f1_zero_cm<<<(N_CLASSES * N_CLASSES + 255) / 256, 256, 0, stream>>>(cm);
  f1_confusion_kernel<<<MAIN_BLOCKS, 256, 0, stream>>>(y_pred, y_true, cm);
  f1_finalize<<<1, N_CLASSES, 0, stream>>>(cm, out);
}